// FingerprintsModel_94489280783
// MI455X (gfx1250) — compile-verified
//
#include <hip/hip_runtime.h>
#include <hip/hip_bf16.h>

typedef float v2f __attribute__((ext_vector_type(2)));
typedef float v8f __attribute__((ext_vector_type(8)));
typedef unsigned int v4u __attribute__((ext_vector_type(4)));
typedef int v8i __attribute__((ext_vector_type(8)));
typedef int v4i __attribute__((ext_vector_type(4)));

#define NNODES 200000
#define NEDGES 800000
#define FIN    128
#define NGRAPH 8000
#define EPSV   1e-5f
#define LDSW_ELEMS 8192   // 32 KB of f32 weight staging per block

// ---------------------------------------------------------------- utilities
__global__ __launch_bounds__(256) void fill_f32(float* __restrict__ p, float v, int n) {
    int i = blockIdx.x * blockDim.x + threadIdx.x;
    if (i < n) p[i] = v;
}

__global__ __launch_bounds__(256) void relu_k(float* __restrict__ p, int n) {
    int i = blockIdx.x * blockDim.x + threadIdx.x;
    if (i < n) p[i] = fmaxf(p[i], 0.0f);
}

// ------------------------------------------------------------------ degrees
__global__ __launch_bounds__(256) void deg_accum(const int* __restrict__ dst,
                                                 float* __restrict__ deg, int E) {
    int e = blockIdx.x * blockDim.x + threadIdx.x;
    if (e < E) unsafeAtomicAdd(&deg[dst[e]], 1.0f);
}

__global__ __launch_bounds__(256) void rsqrt_k(const float* __restrict__ deg,
                                               float* __restrict__ dis, int n) {
    int i = blockIdx.x * blockDim.x + threadIdx.x;
    if (i < n) dis[i] = rsqrtf(deg[i]);
}

// -------------------------------------------------------------- instance norm
__global__ __launch_bounds__(256) void in_accum(const float* __restrict__ x,
                                                const int* __restrict__ batch,
                                                float* __restrict__ sum,
                                                float* __restrict__ sq,
                                                float* __restrict__ cnt, int N) {
    int tid = blockIdx.x * blockDim.x + threadIdx.x;
    int n = tid >> 5;
    int c4 = (tid & 31) << 2;
    if (n >= N) return;
    int g = batch[n];
    float4 v = *(const float4*)(x + (size_t)n * FIN + c4);
    float* s = sum + (size_t)g * FIN + c4;
    float* q = sq  + (size_t)g * FIN + c4;
    unsafeAtomicAdd(s + 0, v.x); unsafeAtomicAdd(q + 0, v.x * v.x);
    unsafeAtomicAdd(s + 1, v.y); unsafeAtomicAdd(q + 1, v.y * v.y);
    unsafeAtomicAdd(s + 2, v.z); unsafeAtomicAdd(q + 2, v.z * v.z);
    unsafeAtomicAdd(s + 3, v.w); unsafeAtomicAdd(q + 3, v.w * v.w);
    if (c4 == 0) unsafeAtomicAdd(&cnt[g], 1.0f);
}

__global__ __launch_bounds__(256) void in_finalize(float* __restrict__ sum,
                                                   float* __restrict__ sq,
                                                   const float* __restrict__ cnt, int GC) {
    int i = blockIdx.x * blockDim.x + threadIdx.x;
    if (i >= GC) return;
    float c = fmaxf(cnt[i >> 7], 1.0f);
    float m = sum[i] / c;
    float var = sq[i] / c - m * m;
    sum[i] = m;                       // becomes mean
    sq[i]  = rsqrtf(var + EPSV);      // becomes inv-std
}

__global__ __launch_bounds__(256) void in_apply(const float* __restrict__ x,
                                                const int* __restrict__ batch,
                                                const float* __restrict__ mean,
                                                const float* __restrict__ istd,
                                                const float* __restrict__ wp,
                                                const float* __restrict__ bp,
                                                float* __restrict__ xn, int N) {
    int tid = blockIdx.x * blockDim.x + threadIdx.x;
    int n = tid >> 7;
    int c = tid & 127;
    if (n >= N) return;
    int g = batch[n];
    float v = x[(size_t)n * FIN + c];
    float m = mean[(size_t)g * FIN + c];
    float is = istd[(size_t)g * FIN + c];
    xn[(size_t)n * FIN + c] = (v - m) * is * wp[0] + bp[0];
}

// --------------------------------------------------- TDM helper (device side)
// Issue TENSOR_LOAD_TO_LDS of `elems` consecutive f32 from `gaddr` into LDS
// byte address `lds_byte`. D# layout per CDNA5 ISA ch.8: group0 = count/lds/
// global/type, group1 = data_size + 1-D tile (tile_dim0 = elems <= 65535),
// groups 2/3 zeroed (dims beyond 2D unused).
__device__ __forceinline__ void tdm_load_f32(unsigned long long gaddr,
                                             unsigned lds_byte, unsigned elems) {
    v4u g0;
    g0[0] = 1u;                                            // count=1, user mode
    g0[1] = lds_byte;                                      // lds_addr (bytes)
    g0[2] = (unsigned)(gaddr & 0xFFFFFFFFull);             // global_addr[31:0]
    g0[3] = (unsigned)((gaddr >> 32) & 0x01FFFFFFull)      // global_addr[56:32]
            | (2u << 30);                                  // type = 2 (image)
    v8i g1;
    g1[0] = 0x20000;                                       // data_size=2 (4B), mask=0
    g1[1] = (int)((elems & 0xFFFFu) << 16);                // tensor_dim0[15:0]
    g1[2] = (int)((elems >> 16) | (1u << 16));             // tensor_dim0[31:16], dim1=1
    g1[3] = (int)(elems << 16);                            // tile_dim0
    g1[4] = 0;                                             // tile_dim1/2 unused
    g1[5] = (int)elems;                                    // tensor_dim0_stride lo
    g1[6] = 0;
    g1[7] = 0;
    v4i z4 = {0, 0, 0, 0};
#if __clang_major__ >= 23
    v8i z8 = {0, 0, 0, 0, 0, 0, 0, 0};
    __builtin_amdgcn_tensor_load_to_lds(g0, g1, z4, z4, z8, 0);
#else
    __builtin_amdgcn_tensor_load_to_lds(g0, g1, z4, z4, 0);
#endif
}

// ------------------------------------------------------------- WMMA f32 GEMM
// C[M,Ncol] = A[M,K] @ W[K,Ncol] (+bias)(+relu), fp32 end-to-end via
// V_WMMA_F32_16X16X4_F32. One wave owns a 16x16 tile. W is staged into LDS in
// <=32KB K-chunks by the Tensor Data Mover (wave 0 issues, TENSORcnt-waited);
// B-fragments then come from ds_load. chunkK is always a multiple of 16 for
// every GEMM in this model (K,Ncol in {32..256} powers of two), so the chunk
// loop is hard-unrolled 4x: 4 back-to-back v_wmma per iteration.
__global__ __launch_bounds__(256) void gemm_wmma_f32(const float* __restrict__ A,
                                                     const float* __restrict__ W,
                                                     const float* __restrict__ bias,
                                                     float* __restrict__ C,
                                                     int M, int K, int Ncol, int flags) {
    __shared__ float ldsW[LDSW_ELEMS];

    int wave = (blockIdx.x * blockDim.x + threadIdx.x) >> 5;
    int lane = threadIdx.x & 31;
    int tilesN = Ncol >> 4;
    int tm = wave / tilesN;
    int tn = wave - tm * tilesN;
    bool active = (tm * 16 < M);
    int m0 = tm << 4, n0 = tn << 4;
    int half = lane >> 4;       // 0: lanes 0-15, 1: lanes 16-31
    int l = lane & 15;

    const float* arow = active ? (A + (size_t)(m0 + l) * K) : A;
    unsigned lds_base = (unsigned)(size_t)(&ldsW[0]);

    int chunkK = LDSW_ELEMS / Ncol;
    if (chunkK > K) chunkK = K;

    v8f acc = {};
    for (int k0 = 0; k0 < K; k0 += chunkK) {
        __syncthreads();        // previous chunk fully consumed
        if (threadIdx.x < 32) { // wave 0 DMAs the W chunk into LDS
            unsigned elems = (unsigned)(chunkK * Ncol);
            unsigned long long ga =
                (unsigned long long)(size_t)W + (unsigned long long)k0 * Ncol * 4ull;
            tdm_load_f32(ga, lds_base, elems);
            __builtin_amdgcn_s_wait_tensorcnt(0);
        }
        __syncthreads();        // chunk visible to all waves

        if (active) {
            for (int kk = 0; kk < chunkK; kk += 16) {
#pragma unroll
                for (int u = 0; u < 4; ++u) {
                    int ka = kk + 4 * u + 2 * half;
                    v2f a, b;
                    a.x = arow[k0 + ka];
                    a.y = arow[k0 + ka + 1];
                    b.x = ldsW[(size_t)ka * Ncol + n0 + l];
                    b.y = ldsW[(size_t)(ka + 1) * Ncol + n0 + l];
                    acc = __builtin_amdgcn_wmma_f32_16x16x4_f32(
                        false, a, false, b, (short)0, acc, false, false);
                }
            }
        }
    }

    if (active) {
        float bv = (flags & 1) ? bias[n0 + l] : 0.0f;
        int rbase = m0 + (half ? 8 : 0);
#pragma unroll
        for (int i = 0; i < 8; ++i) {
            float v = acc[i] + bv;
            if (flags & 2) v = fmaxf(v, 0.0f);
            C[(size_t)(rbase + i) * Ncol + n0 + l] = v;
        }
    }
}

// ----------------------------------------------------------- GCN aggregation
__global__ __launch_bounds__(256) void agg_init(const float* __restrict__ m,
                                                const float* __restrict__ dis,
                                                const float* __restrict__ bias,
                                                float* __restrict__ out, int N, int C) {
    int tid = blockIdx.x * blockDim.x + threadIdx.x;
    if (tid >= N * C) return;
    int n = tid / C;
    int c = tid - n * C;
    float d = dis[n];
    out[tid] = m[tid] * d * d + bias[c];
}

__global__ __launch_bounds__(256) void agg_edges(const float* __restrict__ m,
                                                 const int* __restrict__ src,
                                                 const int* __restrict__ dst,
                                                 const float* __restrict__ dis,
                                                 float* __restrict__ out, int E, int C) {
    int tid = blockIdx.x * blockDim.x + threadIdx.x;
    int c4chunks = C >> 2;
    int e = tid / c4chunks;
    if (e >= E) return;
    int c4 = (tid - e * c4chunks) << 2;
    int s = src[e], d = dst[e];
    float w = dis[s] * dis[d];
    float4 v = *(const float4*)(m + (size_t)s * C + c4);
    float* o = out + (size_t)d * C + c4;
    unsafeAtomicAdd(o + 0, v.x * w);
    unsafeAtomicAdd(o + 1, v.y * w);
    unsafeAtomicAdd(o + 2, v.z * w);
    unsafeAtomicAdd(o + 3, v.w * w);
}

// ------------------------------------------------------------------- pooling
__global__ __launch_bounds__(256) void pool_max(const float* __restrict__ h,
                                                const int* __restrict__ batch,
                                                float* __restrict__ pooled, int N) {
    int tid = blockIdx.x * blockDim.x + threadIdx.x;
    int n = tid >> 7;
    int c = tid & 127;
    if (n >= N) return;
    int g = batch[n];
    float v = h[(size_t)n * 128 + c];
    atomicMax((int*)(pooled + (size_t)g * 128 + c), __float_as_int(v));
}

// ---------------------------------------------------------------- launchers
static inline int cdiv(long long a, long long b) { return (int)((a + b - 1) / b); }

static void launch_gemm(const float* A, const float* W, const float* bias, float* C,
                        int M, int K, int Ncol, int flags, hipStream_t s) {
    long long waves = (long long)(M / 16) * (Ncol / 16);
    long long threads = waves * 32;
    gemm_wmma_f32<<<cdiv(threads, 256), 256, 0, s>>>(A, W, bias, C, M, K, Ncol, flags);
}

extern "C" void kernel_launch(void* const* d_in, const int* in_sizes, int n_in,
                              void* d_out, int out_size, void* d_ws, size_t ws_size,
                              hipStream_t stream) {
    const float* x       = (const float*)d_in[0];
    const int*   eidx    = (const int*)d_in[1];     // [2,E]: src then dst
    const int*   batch   = (const int*)d_in[2];
    const float* norm_w  = (const float*)d_in[3];
    const float* norm_b  = (const float*)d_in[4];
    const float* W1      = (const float*)d_in[5];
    const float* b1      = (const float*)d_in[6];
    const float* W2      = (const float*)d_in[7];
    const float* b2      = (const float*)d_in[8];
    const float* W3      = (const float*)d_in[9];
    const float* b3      = (const float*)d_in[10];
    const float* lin1_W  = (const float*)d_in[11];
    const float* lin1_b  = (const float*)d_in[12];
    const float* lin2_W  = (const float*)d_in[13];
    const float* lin2_b  = (const float*)d_in[14];
    float* out = (float*)d_out;

    const int N = NNODES, E = NEDGES, G = NGRAPH;
    const int* src = eidx;
    const int* dst = eidx + E;

    // ---- workspace carve-up (floats)
    float* ws = (float*)d_ws;
    size_t off = 0;
    float* deg   = ws + off; off += N;
    float* dis   = ws + off; off += N;
    float* cnt   = ws + off; off += G;
    float* mean  = ws + off; off += (size_t)G * 128;   // sum -> mean
    float* istd  = ws + off; off += (size_t)G * 128;   // sumsq -> inv std
    float* R0    = ws + off; off += (size_t)N * 128;   // xn, later a3
    float* R1    = ws + off; off += (size_t)N * 128;   // m1 / m2 / m3
    float* A1    = ws + off; off += (size_t)N * 32;
    float* A2    = ws + off; off += (size_t)N * 64;
    float* pool  = ws + off; off += (size_t)G * 128;
    float* l1out = ws + off; off += (size_t)G * 256;
    (void)ws_size; (void)n_in; (void)in_sizes; (void)out_size;

    // ---- degrees: deg = 1 + segsum(1 over dst); dis = rsqrt(deg)
    fill_f32<<<cdiv(N, 256), 256, 0, stream>>>(deg, 1.0f, N);
    deg_accum<<<cdiv(E, 256), 256, 0, stream>>>(dst, deg, E);
    rsqrt_k<<<cdiv(N, 256), 256, 0, stream>>>(deg, dis, N);

    // ---- instance norm
    fill_f32<<<cdiv(G * 128, 256), 256, 0, stream>>>(mean, 0.0f, G * 128);
    fill_f32<<<cdiv(G * 128, 256), 256, 0, stream>>>(istd, 0.0f, G * 128);
    fill_f32<<<cdiv(G, 256), 256, 0, stream>>>(cnt, 0.0f, G);
    in_accum<<<cdiv((long long)N * 32, 256), 256, 0, stream>>>(x, batch, mean, istd, cnt, N);
    in_finalize<<<cdiv(G * 128, 256), 256, 0, stream>>>(mean, istd, cnt, G * 128);
    in_apply<<<cdiv((long long)N * 128, 256), 256, 0, stream>>>(x, batch, mean, istd,
                                                                norm_w, norm_b, R0, N);

    // ---- conv1: m1 = xn@W1 ; a1 = scatter + self + b1 ; relu
    launch_gemm(R0, W1, nullptr, R1, N, 128, 32, 0, stream);
    agg_init<<<cdiv((long long)N * 32, 256), 256, 0, stream>>>(R1, dis, b1, A1, N, 32);
    agg_edges<<<cdiv((long long)E * 8, 256), 256, 0, stream>>>(R1, src, dst, dis, A1, E, 32);
    relu_k<<<cdiv((long long)N * 32, 256), 256, 0, stream>>>(A1, N * 32);

    // ---- conv2
    launch_gemm(A1, W2, nullptr, R1, N, 32, 64, 0, stream);
    agg_init<<<cdiv((long long)N * 64, 256), 256, 0, stream>>>(R1, dis, b2, A2, N, 64);
    agg_edges<<<cdiv((long long)E * 16, 256), 256, 0, stream>>>(R1, src, dst, dis, A2, E, 64);
    relu_k<<<cdiv((long long)N * 64, 256), 256, 0, stream>>>(A2, N * 64);

    // ---- conv3 (a3 overwrites xn region R0)
    launch_gemm(A2, W3, nullptr, R1, N, 64, 128, 0, stream);
    agg_init<<<cdiv((long long)N * 128, 256), 256, 0, stream>>>(R1, dis, b3, R0, N, 128);
    agg_edges<<<cdiv((long long)E * 32, 256), 256, 0, stream>>>(R1, src, dst, dis, R0, E, 128);
    relu_k<<<cdiv((long long)N * 128, 256), 256, 0, stream>>>(R0, N * 128);

    // ---- global max pool (values >= 0, zero init == reference's clamp to 0)
    fill_f32<<<cdiv(G * 128, 256), 256, 0, stream>>>(pool, 0.0f, G * 128);
    pool_max<<<cdiv((long long)N * 128, 256), 256, 0, stream>>>(R0, batch, pool, N);

    // ---- MLP head: lin1 (+bias,+relu), lin2 (+bias)
    launch_gemm(pool, lin1_W, lin1_b, l1out, G, 128, 256, 3, stream);
    launch_gemm(l1out, lin2_W, lin2_b, out, G, 256, 128, 1, stream);
}